// SimplicialKuramotoLayer_30202210025981
// MI455X (gfx1250) — compile-verified
//
#include <hip/hip_runtime.h>
#include <stdint.h>

typedef __attribute__((ext_vector_type(2))) float v2f;
typedef __attribute__((ext_vector_type(8))) float v8f;

#define KN 8192
#define KN_STEPS 50
#define KDT 0.01f

#define ROWS_PER_BLOCK 128            // 8 waves x 16 rows
#define CCHUNKS 4                     // column split: 4x more blocks in flight
#define CCOLS (KN / CCHUNKS)          // 2048 columns per block
#define TJ 32                         // columns per staged tile
#define TILE_STRIDE (TJ + 4)          // 36 floats: rows stay 16B aligned, b64 reads conflict-free
#define NTILES (CCOLS / TJ)           // 64 tiles per block
#define JSLOTS (TJ / 4)               // 8 WMMAs per tile per wave

// ---------------------------------------------------------------------------
// CDNA5 async global -> LDS copy (bypasses VGPRs, tracked by ASYNCcnt)
// ---------------------------------------------------------------------------
__device__ __forceinline__ void async_load_b128(uint32_t lds_addr, const float* gaddr) {
  asm volatile("global_load_async_to_lds_b128 %0, %1, off"
               :: "v"(lds_addr), "v"(gaddr) : "memory");
}

// ---------------------------------------------------------------------------
// Kernel 1: s = sin(theta), c = cos(theta), per-block partial sums
// Grid: KN/256 = 32 blocks of 256 threads
// ---------------------------------------------------------------------------
__global__ void kuramoto_phase_kernel(const float* __restrict__ theta,
                                      float* __restrict__ s,
                                      float* __restrict__ c,
                                      float* __restrict__ pS,
                                      float* __restrict__ pC) {
  __shared__ float wS[8], wC[8];
  int i = blockIdx.x * 256 + threadIdx.x;
  float th = theta[i];
  float sv = sinf(th);
  float cv = cosf(th);
  s[i] = sv;
  c[i] = cv;
  for (int o = 16; o > 0; o >>= 1) {
    sv += __shfl_down(sv, o, 32);
    cv += __shfl_down(cv, o, 32);
  }
  int lane = threadIdx.x & 31;
  int wave = threadIdx.x >> 5;
  if (lane == 0) { wS[wave] = sv; wC[wave] = cv; }
  __syncthreads();
  if (threadIdx.x == 0) {
    float a = 0.f, b = 0.f;
#pragma unroll
    for (int w = 0; w < 8; ++w) { a += wS[w]; b += wC[w]; }
    pS[blockIdx.x] = a;
    pC[blockIdx.x] = b;
  }
}

// ---------------------------------------------------------------------------
// Kernel 2: partial dual matvec y1 += K@s, y2 += K@c over a 128-row x
// 2048-column chunk, via V_WMMA_F32_16X16X4_F32 with async-to-LDS double
// buffering. Grid: 64 row-blocks x 4 column-chunks = 256 blocks of 256.
// ---------------------------------------------------------------------------
__global__ void kuramoto_matvec_kernel(const float* __restrict__ Kmat,
                                       const float* __restrict__ s,
                                       const float* __restrict__ c,
                                       float* __restrict__ y1p,   // [CCHUNKS][KN]
                                       float* __restrict__ y2p) { // [CCHUNKS][KN]
  __shared__ float Kbuf[2][ROWS_PER_BLOCK * TILE_STRIDE]; // 2 x 18 KB
  __shared__ float btile[2][JSLOTS * 64];                 // 2 x 2 KB B fragments

  const int tid  = threadIdx.x;
  const int lane = tid & 31;
  const int wave = tid >> 5;
  const int half = lane >> 4;   // A layout: 0 -> K=0,1 ; 1 -> K=2,3
  const int mrow = lane & 15;   // M within the wave's 16-row tile

  const int rowblk   = blockIdx.x & 63;
  const int cchunk   = blockIdx.x >> 6;
  const int rowbase  = rowblk * ROWS_PER_BLOCK;
  const int colbase  = cchunk * CCOLS;
  const int waveRow0 = wave * 16;

  // ---- per-thread addresses for the async K-tile stream ----
  // Each tile: 128 rows x 8 float4 = 1024 b128 transfers -> 4 per thread.
  const int r0 = tid >> 3;   // rows r0, r0+32, r0+64, r0+96
  const int cq = tid & 7;    // float4 column within row
  const float* gRow[4];
  uint32_t ldsDst[2][4];
#pragma unroll
  for (int k = 0; k < 4; ++k) {
    gRow[k] = Kmat + (size_t)(rowbase + r0 + 32 * k) * KN + colbase + cq * 4;
    ldsDst[0][k] = (uint32_t)(uintptr_t)&Kbuf[0][(r0 + 32 * k) * TILE_STRIDE + cq * 4];
    ldsDst[1][k] = (uint32_t)(uintptr_t)&Kbuf[1][(r0 + 32 * k) * TILE_STRIDE + cq * 4];
  }

  // ---- B-fragment slot mapping (only lanes 0,1,16,17 are nonzero) ----
  const int bq    = tid >> 3;
  const int br    = tid & 7;
  const int blane = ((br & 4) << 2) | (br & 1);           // 0,1,16,17
  const int bcomp = (br >> 1) & 1;
  const int bcol  = 4 * bq + ((br & 4) ? 2 : 0) + bcomp;  // K-column in chunk
  const int bdst  = bq * 64 + blane * 2 + bcomp;
  const float* bsrc = ((br & 1) ? c : s) + colbase;       // lane 0/16 <- s, 1/17 <- c

  // zero both btile buffers once (zero lanes never rewritten)
  for (int i = tid; i < 2 * JSLOTS * 64; i += 256) (&btile[0][0])[i] = 0.f;
  __syncthreads();

  v8f acc = {}; // C tile: column 0 accumulates K@s, column 1 accumulates K@c

  // ---- prologue: stage tile 0 ----
  if (tid < 64) btile[0][bdst] = bsrc[bcol];
#pragma unroll
  for (int k = 0; k < 4; ++k) async_load_b128(ldsDst[0][k], gRow[k]);

  for (int t = 0; t < NTILES; ++t) {
    const int cur = t & 1;
    if (t + 1 < NTILES) {
      const int jb1 = (t + 1) * TJ;
      if (tid < 64) btile[cur ^ 1][bdst] = bsrc[jb1 + bcol];
#pragma unroll
      for (int k = 0; k < 4; ++k) async_load_b128(ldsDst[cur ^ 1][k], gRow[k] + jb1);
      // batch for tile t was issued first; ASYNCcnt is in-order -> <=4 means done
      asm volatile("s_wait_asynccnt 0x4" ::: "memory");
    } else {
      asm volatile("s_wait_asynccnt 0x0" ::: "memory");
    }
    __syncthreads(); // publish Kbuf[cur] + btile[cur] across waves

#pragma unroll
    for (int q = 0; q < JSLOTS; ++q) {
      v2f a = *reinterpret_cast<const v2f*>(
          &Kbuf[cur][(waveRow0 + mrow) * TILE_STRIDE + 4 * q + half * 2]);
      v2f b = *reinterpret_cast<const v2f*>(&btile[cur][q * 64 + lane * 2]);
      acc = __builtin_amdgcn_wmma_f32_16x16x4_f32(
          false, a, false, b, (short)0, acc, false, false);
    }
    __syncthreads(); // readers done before next iteration overwrites this buffer
  }

  // ---- store partials: lane N=0 holds K@s rows, lane N=1 holds K@c rows ----
  // C/D layout: VGPR i <-> M=i (lanes 0-15) or M=i+8 (lanes 16-31); lane = N.
  if (mrow < 2) {
    float* dst = (mrow == 0 ? y1p : y2p) + (size_t)cchunk * KN;
#pragma unroll
    for (int i = 0; i < 8; ++i) {
      int row = rowbase + waveRow0 + half * 8 + i;
      dst[row] = acc[i];
    }
  }
}

// ---------------------------------------------------------------------------
// Kernel 3: combine column-chunk partials, reduce S/C, fused Euler update.
// Grid: KN/256 = 32 blocks of 256 threads.
// ---------------------------------------------------------------------------
__global__ void kuramoto_finalize_kernel(float* __restrict__ theta,
                                         const float* __restrict__ omegas,
                                         const float* __restrict__ s,
                                         const float* __restrict__ c,
                                         const float* __restrict__ y1p,
                                         const float* __restrict__ y2p,
                                         const float* __restrict__ pS,
                                         const float* __restrict__ pC,
                                         const float* __restrict__ sigma2p) {
  __shared__ float bc[2];
  const int tid  = threadIdx.x;
  const int lane = tid & 31;
  if (tid < 32) {
    float v = pS[lane];
    for (int o = 16; o > 0; o >>= 1) v += __shfl_down(v, o, 32);
    if (lane == 0) bc[0] = v;
  } else if (tid < 64) {
    float v = pC[lane];
    for (int o = 16; o > 0; o >>= 1) v += __shfl_down(v, o, 32);
    if (lane == 0) bc[1] = v;
  }
  __syncthreads();

  const int i = blockIdx.x * 256 + tid;
  float y1 = 0.f, y2 = 0.f;
#pragma unroll
  for (int k = 0; k < CCHUNKS; ++k) {
    y1 += y1p[(size_t)k * KN + i];
    y2 += y2p[(size_t)k * KN + i];
  }

  const float Ssum = bc[0];
  const float Csum = bc[1];
  const float sigma2 = *sigma2p;
  const float inv_n  = 1.0f / (float)KN;
  const float inv_n2 = inv_n * inv_n;

  float th   = theta[i];
  float pair = (c[i] * y1 - s[i] * y2) * inv_n;
  float c2   = cosf(2.0f * th);
  float s2   = sinf(2.0f * th);
  float tri  = sigma2 * (2.0f * Csum * Ssum * c2 -
                         (Csum * Csum - Ssum * Ssum) * s2) * inv_n2;
  theta[i] = th + KDT * (omegas[i] + pair + tri);
}

// ---------------------------------------------------------------------------
__global__ void kuramoto_copy_kernel(const float* __restrict__ src,
                                     float* __restrict__ dst) {
  int i = blockIdx.x * 256 + threadIdx.x;
  dst[i] = src[i];
}

// ---------------------------------------------------------------------------
extern "C" void kernel_launch(void* const* d_in, const int* in_sizes, int n_in,
                              void* d_out, int out_size, void* d_ws, size_t ws_size,
                              hipStream_t stream) {
  const float* phases = (const float*)d_in[0];
  const float* Kmat   = (const float*)d_in[1];
  const float* omegas = (const float*)d_in[2];
  const float* sigma2 = (const float*)d_in[3];

  float* ws    = (float*)d_ws;
  float* theta = ws;                  // KN
  float* s     = ws + KN;             // KN
  float* c     = ws + 2 * KN;         // KN
  float* pS    = ws + 3 * KN;         // 32 (padded to 64)
  float* pC    = ws + 3 * KN + 64;    // 32 (padded to 64)
  float* y1p   = ws + 3 * KN + 128;               // CCHUNKS*KN partials
  float* y2p   = ws + 3 * KN + 128 + CCHUNKS * KN;

  // Re-init theta from the immutable input every call (graph-replay safe).
  kuramoto_copy_kernel<<<KN / 256, 256, 0, stream>>>(phases, theta);

  for (int step = 0; step < KN_STEPS; ++step) {
    kuramoto_phase_kernel<<<KN / 256, 256, 0, stream>>>(theta, s, c, pS, pC);
    kuramoto_matvec_kernel<<<64 * CCHUNKS, 256, 0, stream>>>(Kmat, s, c, y1p, y2p);
    kuramoto_finalize_kernel<<<KN / 256, 256, 0, stream>>>(
        theta, omegas, s, c, y1p, y2p, pS, pC, sigma2);
  }

  kuramoto_copy_kernel<<<KN / 256, 256, 0, stream>>>(theta, (float*)d_out);
}